// Actor_44135083933697
// MI455X (gfx1250) — compile-verified
//
#include <hip/hip_runtime.h>
#include <math.h>

typedef float v2f __attribute__((ext_vector_type(2)));
typedef float v8f __attribute__((ext_vector_type(8)));

// Exact parameter type revealed by hipcc: int __vector_size__(16), AS1/AS3 ptrs.
typedef int gi4 __attribute__((vector_size(4 * sizeof(int))));
typedef __attribute__((address_space(1))) gi4 gi4_global;
typedef __attribute__((address_space(3))) gi4 gi4_lds;

#define B_DIM 64
#define L_DIM 2048
#define H_DIM 128
#define O_DIM 128
#define N_DIM 256

#define LCHUNK 128           // l-range per block (grid.z = L/LCHUNK = 16)
#define WAVES  4             // waves per block; each wave owns one l per stage
#define TPB    (WAVES * 32)
#define STAGES (LCHUNK / WAVES)          // 32
#define F4_PER_STAGE (WAVES * 16 * (H_DIM / 4))   // 2048 float4 per stage
#define ITERS  (F4_PER_STAGE / TPB)               // 16 b128 ops per thread/stage

#if defined(__HIP_DEVICE_COMPILE__) && __has_builtin(__builtin_amdgcn_global_load_async_to_lds_b128)
#define HAVE_ASYNC_LDS 1
#else
#define HAVE_ASYNC_LDS 0
#endif

// ---- async global->LDS b128 (gfx1250), with safe synchronous fallback ----
__device__ __forceinline__ void copy_b128(const float* g, float* l) {
#if HAVE_ASYNC_LDS
    __builtin_amdgcn_global_load_async_to_lds_b128(
        (gi4_global*)g, (gi4_lds*)l, 0, 0);
#else
    *(float4*)l = *(const float4*)g;
#endif
}

template <int N>
__device__ __forceinline__ void wait_async() {
#if HAVE_ASYNC_LDS
#if __has_builtin(__builtin_amdgcn_s_wait_asynccnt)
    __builtin_amdgcn_s_wait_asynccnt(N);
#else
    asm volatile("s_wait_asynccnt %0" ::"i"(N) : "memory");
#endif
#endif
}

// ---------------------------------------------------------------- zero ws
__global__ void zero_kernel(float* __restrict__ p, int n) {
    int i = blockIdx.x * blockDim.x + threadIdx.x;
    if (i < n) p[i] = 0.0f;
}

// ------------------------------------------- weights w[l,n] = gamma*lam^k
__global__ void weight_kernel(const float* __restrict__ nu_log,
                              const float* __restrict__ theta_log,
                              const float* __restrict__ gamma_log,
                              float* __restrict__ Wre,
                              float* __restrict__ Wim) {
    int idx = blockIdx.x * blockDim.x + threadIdx.x;
    if (idx >= L_DIM * N_DIM) return;
    int l = idx / N_DIM;
    int n = idx - l * N_DIM;
    float k     = (float)(L_DIM - 1 - l);
    float e_nu  = __expf(nu_log[n]);     // |lam| = exp(-e_nu)
    float theta = __expf(theta_log[n]);  // arg(lam)
    float g     = __expf(gamma_log[n]);  // gamma scale folded into weight
    float mag   = __expf(-k * e_nu) * g;
    float ang   = fmodf(k * theta, 6.28318530717958647692f);
    float s, c;
    __sincosf(ang, &s, &c);
    Wre[idx] = mag * c;
    Wim[idx] = mag * s;
}

// ---------------------------------------------------------- main WMMA GEMM
// Per block: 16 b-rows x 16 n-cols tile, reduced over LCHUNK l's and H=128.
// Double-buffered LDS staging driven by async global->LDS loads.
__global__ __launch_bounds__(TPB) void ssm_wmma_kernel(
    const float* __restrict__ u,      // (B, L, H)
    const float* __restrict__ Bre,    // (N, H)
    const float* __restrict__ Bim,    // (N, H)
    const float* __restrict__ Wre,    // (L, N)
    const float* __restrict__ Wim,    // (L, N)
    float* __restrict__ xr_g,         // (B, N) accumulators
    float* __restrict__ xi_g) {
    __shared__ __align__(16) float su[2][WAVES][16][132];  // 132: bank pad

    const int n0    = blockIdx.x * 16;
    const int b0    = blockIdx.y * 16;
    const int lroot = blockIdx.z * LCHUNK;
    const int tid   = threadIdx.x;
    const int w     = tid >> 5;       // wave id
    const int lane  = tid & 31;
    const int half  = lane >> 4;      // K-half selector (K=2*half .. 2*half+1)
    const int sub   = lane & 15;      // A: row m ; B/C/D: col n

    // B_re / B_im fragments resident in VGPRs for the whole block.
    v2f brf[32], bif[32];
    {
        const v2f* brp = (const v2f*)(Bre + (size_t)(n0 + sub) * H_DIM);
        const v2f* bip = (const v2f*)(Bim + (size_t)(n0 + sub) * H_DIM);
#pragma unroll
        for (int kb = 0; kb < 32; ++kb) {       // kb*4 = h base
            brf[kb] = brp[2 * kb + half];
            bif[kb] = bip[2 * kb + half];
        }
    }

    // Per-thread (li, m, h4) slot for cooperative staging.
    // idx = it*TPB + tid over 2048 float4s: li = idx/512, m = (idx/32)%16, h4 = idx%32.
    auto issue_stage = [&](int s, int buf) {
        const int lb = lroot + s * WAVES;
#pragma unroll
        for (int it = 0; it < ITERS; ++it) {
            int idx = it * TPB + tid;
            int li  = idx >> 9;
            int rem = idx & 511;
            int m   = rem >> 5;
            int h4  = rem & 31;
            const float* g =
                u + ((size_t)(b0 + m) * L_DIM + (size_t)(lb + li)) * H_DIM + 4 * h4;
            copy_b128(g, &su[buf][li][m][4 * h4]);   // b128 async -> LDS
        }
    };

    v8f xr = {}, xi = {};

    issue_stage(0, 0);                                // prologue: stage 0
    for (int s = 0; s < STAGES; ++s) {
        if (s + 1 < STAGES) {
            issue_stage(s + 1, (s + 1) & 1);          // overlap next stage's HBM
            wait_async<ITERS>();                      // stage s complete (in-order)
        } else {
            wait_async<0>();
        }
        __syncthreads();                              // stage-s LDS visible to all

        // This wave consumes l = lb + w.
        {
            const int l  = lroot + s * WAVES + w;
            const float wr = Wre[(size_t)l * N_DIM + n0 + sub];
            const float wi = Wim[(size_t)l * N_DIM + n0 + sub];
            v8f P = {}, Q = {};
            const v2f* ap = (const v2f*)&su[s & 1][w][sub][0];
#pragma unroll
            for (int kb = 0; kb < 32; ++kb) {
                v2f a = ap[2 * kb + half];            // ds_load_b64
                P = __builtin_amdgcn_wmma_f32_16x16x4_f32(
                        false, a, false, brf[kb], (short)0, P, false, false);
                Q = __builtin_amdgcn_wmma_f32_16x16x4_f32(
                        false, a, false, bif[kb], (short)0, Q, false, false);
            }
            // x += (wr + i*wi) * (P + i*Q)
#pragma unroll
            for (int v = 0; v < 8; ++v) {
                xr[v] = fmaf(wr, P[v], fmaf(-wi, Q[v], xr[v]));
                xi[v] = fmaf(wi, P[v], fmaf( wr, Q[v], xi[v]));
            }
        }
        __syncthreads();                              // buffer consumed by all waves
    }

    // Accumulate partial tile: C/D layout -> row b0 + v + 8*half, col n0 + sub.
#pragma unroll
    for (int v = 0; v < 8; ++v) {
        int b = b0 + v + 8 * half;
        int n = n0 + sub;
        unsafeAtomicAdd(&xr_g[(size_t)b * N_DIM + n], xr[v]);
        unsafeAtomicAdd(&xi_g[(size_t)b * N_DIM + n], xi[v]);
    }
}

// ------------------------------------------------------------- epilogue
__global__ void out_kernel(const float* __restrict__ xr,
                           const float* __restrict__ xi,
                           const float* __restrict__ Cre,   // (O, N)
                           const float* __restrict__ Cim,   // (O, N)
                           const float* __restrict__ D,     // (O, H)
                           const float* __restrict__ u,     // (B, L, H)
                           float* __restrict__ out) {       // (B, O)
    int b = blockIdx.x;
    int o = threadIdx.x;
    float acc = 0.0f;
    for (int n = 0; n < N_DIM; ++n)
        acc = fmaf(Cre[o * N_DIM + n],  xr[b * N_DIM + n],
              fmaf(-Cim[o * N_DIM + n], xi[b * N_DIM + n], acc));
    const float* ul = u + ((size_t)b * L_DIM + (L_DIM - 1)) * H_DIM;
    for (int h = 0; h < H_DIM; ++h)
        acc = fmaf(D[o * H_DIM + h], ul[h], acc);
    out[(size_t)b * O_DIM + o] = acc;
}

// ---------------------------------------------------------------- launch
extern "C" void kernel_launch(void* const* d_in, const int* in_sizes, int n_in,
                              void* d_out, int out_size, void* d_ws, size_t ws_size,
                              hipStream_t stream) {
    // setup_inputs order:
    // 0 state (unused), 1 dynamics_input_time_window (unused),
    // 2 dynamics_disturbance_time_window, 3 nu_log, 4 theta_log, 5 gamma_log,
    // 6 B_re, 7 B_im, 8 C_re, 9 C_im, 10 D
    const float* u      = (const float*)d_in[2];
    const float* nu_log = (const float*)d_in[3];
    const float* th_log = (const float*)d_in[4];
    const float* ga_log = (const float*)d_in[5];
    const float* Bre    = (const float*)d_in[6];
    const float* Bim    = (const float*)d_in[7];
    const float* Cre    = (const float*)d_in[8];
    const float* Cim    = (const float*)d_in[9];
    const float* D      = (const float*)d_in[10];
    float* out          = (float*)d_out;

    float* ws  = (float*)d_ws;
    float* Wre = ws;                       // L*N
    float* Wim = Wre + (size_t)L_DIM * N_DIM;
    float* xr  = Wim + (size_t)L_DIM * N_DIM;  // B*N (xr,xi contiguous)
    float* xi  = xr + (size_t)B_DIM * N_DIM;

    int nacc = 2 * B_DIM * N_DIM;
    zero_kernel<<<(nacc + 255) / 256, 256, 0, stream>>>(xr, nacc);

    int nw = L_DIM * N_DIM;
    weight_kernel<<<(nw + 255) / 256, 256, 0, stream>>>(nu_log, th_log, ga_log, Wre, Wim);

    dim3 grid(N_DIM / 16, B_DIM / 16, L_DIM / LCHUNK);   // 16 x 4 x 16 = 1024 blocks
    ssm_wmma_kernel<<<grid, TPB, 0, stream>>>(u, Bre, Bim, Wre, Wim, xr, xi);

    out_kernel<<<B_DIM, O_DIM, 0, stream>>>(xr, xi, Cre, Cim, D, u, out);
}